// QAEReal_9311489098142
// MI455X (gfx1250) — compile-verified
//
#include <hip/hip_runtime.h>
#include <math.h>

// ---------------------------------------------------------------------------
// QAE noisy-circuit simulator for gfx1250 (MI455X).
// Ref qubits never touched -> 256-dim real state per batch element.
// RY layers: S(16x16) <- R_hi * S * R_lo^T via chained V_WMMA_F32_16X16X4_F32.
// CNOT / X / Z / XX / ZZ noise = in-wave LDS permutation gathers + sign flips.
// One wave32 per batch element; state lives in a 1KB private LDS tile.
// PRNG decisions are wave-uniform *integer* compares so they lower to SALU.
// ---------------------------------------------------------------------------

typedef float v2f __attribute__((ext_vector_type(2)));
typedef float v8f __attribute__((ext_vector_type(8)));

#define NLAYERS   7        // M=6 entangling layers + final RY layer
#define MLAY      6
// LEVEL = 0.5 lerps from the reference, as p * 2^32 integer thresholds
#define T_P1Q     11811160u   // 0.00275
#define T_P2Q1Q   24696062u   // 0.00575
#define T_P2QPR   14602889u   // 0.0034
#define T_PIDLE    7086696u   // 0.00165
#define OVERROT_C 0.0125f
#define READOUT_C 0.0125f
#define FDEPOL    0.025f

// ----- deterministic counter-based hash PRNG (stands in for jax threefry) ---
__device__ __forceinline__ unsigned hmix(unsigned a, unsigned b) {
    unsigned h = a * 0x9E3779B9u + b * 0x85EBCA77u + 0x165667B1u;
    h ^= h >> 15; h *= 0x2C1B3C6Du;
    h ^= h >> 12; h *= 0x297A2D39u;
    h ^= h >> 15;
    return h;
}
__device__ __forceinline__ float u01(unsigned a, unsigned b) {
    return (float)(hmix(a, b) >> 8) * (1.0f / 16777216.0f);
}

// ---------------------------------------------------------------------------
// Prep kernel: build the 14 fused 16x16 RY matrices (row-major) into d_ws.
//   Rall[(m*2+0)*256 + r*16 + c] = R_hi(r,c)   (qubits 0..3, qubit0 = MSB)
//   Rall[(m*2+1)*256 + n*16 + k] = R_lo(n,k)   (qubits 4..7) == B(k,n)=R_lo^T
// Over-rotation: scalar Gaussian per gate via Box-Muller on the hash stream.
// ---------------------------------------------------------------------------
__global__ void qae_prep(const float* __restrict__ theta, float* __restrict__ Rall) {
    for (int idx = threadIdx.x; idx < NLAYERS * 2 * 256; idx += blockDim.x) {
        int m     = idx >> 9;
        int which = (idx >> 8) & 1;
        int r     = (idx >> 4) & 15;
        int c     = idx & 15;
        float v = 1.0f;
        for (int k = 0; k < 4; k++) {
            int q = 4 * which + k;
            unsigned gid = (unsigned)(m * 8 + q);
            float u1 = u01(0xA11CE000u + 2u * gid, 0x5EEDu);
            float u2 = u01(0xA11CE001u + 2u * gid, 0x5EEDu);
            u1 = fmaxf(u1, 1e-7f);
            float nrm = sqrtf(-2.0f * __logf(u1)) * __cosf(6.2831853f * u2);
            float ang = theta[m * 8 + q] + OVERROT_C * nrm;
            float c0 = __cosf(0.5f * ang), s0 = __sinf(0.5f * ang);
            int bi = (r >> (3 - k)) & 1;   // qubit k occupies nibble bit (3-k)
            int bj = (c >> (3 - k)) & 1;
            float e = (bi == bj) ? c0 : (bi ? s0 : -s0);  // RY = [[c,-s],[s,c]]
            v *= e;
        }
        Rall[idx] = v;
    }
}

// ---------------------------------------------------------------------------
// Main simulator
// ---------------------------------------------------------------------------
__device__ __forceinline__ v8f wmma4(v2f a, v2f b, v8f c) {
    // D(16x16) += A(16x4) * B(4x16), f32 (V_WMMA_F32_16X16X4_F32)
    return __builtin_amdgcn_wmma_f32_16x16x4_f32(false, a, false, b, (short)0, c,
                                                 false, false);
}

// new[i] = sign(i) * old[i ^ srcxor];  sign = -1 iff dosign && parity(i&signmask)
__device__ __forceinline__ void gather_signed(float* S, int lane, int srcxor,
                                              int signmask, int dosign) {
    float t[8];
#pragma unroll
    for (int k = 0; k < 8; k++) t[k] = S[(lane + 32 * k) ^ srcxor];
#pragma unroll
    for (int k = 0; k < 8; k++) {
        int i = lane + 32 * k;
        float v = t[k];
        v = (dosign && (__popc(i & signmask) & 1)) ? -v : v;
        S[i] = v;
    }
}

// CNOT: amp'[i] = amp[i ^ ((i>>bc & 1) << bt)]  (involution on the index)
__device__ __forceinline__ void apply_cnot(float* S, int lane, int bc, int bt) {
    float t[8];
#pragma unroll
    for (int k = 0; k < 8; k++) {
        int i = lane + 32 * k;
        t[k] = S[i ^ (((i >> bc) & 1) << bt)];
    }
#pragma unroll
    for (int k = 0; k < 8; k++) S[lane + 32 * k] = t[k];
}

// X/Z channel event: wave-uniform integer-threshold draw (lowered to SALU),
// gather executed (full EXEC) only on the ~0.3% of events that fire.
__device__ __forceinline__ unsigned xz_event(float* S, int lane, int bit,
                                             unsigned thr, unsigned ev, unsigned b) {
    unsigned h1 = hmix(ev++, b);
    unsigned h2 = hmix(ev++, b);
    if (h1 < thr) {
        int isx = (h2 >> 31) == 0;             // u2 < 0.5
        gather_signed(S, lane, isx ? (1 << bit) : 0, 1 << bit, !isx);
    }
    return ev;
}

__device__ __forceinline__ unsigned pair_event(float* S, int lane, int mask,
                                               unsigned thr, unsigned ev, unsigned b) {
    unsigned h1 = hmix(ev++, b);
    unsigned h2 = hmix(ev++, b);
    if (h1 < thr) {
        int isx = (h2 >> 31) == 0;
        gather_signed(S, lane, isx ? mask : 0, mask, !isx);
    }
    return ev;
}

__global__ void __launch_bounds__(256) qae_sim(const float* __restrict__ x,
                                               const float* __restrict__ Rall,
                                               const float* __restrict__ beta,
                                               const float* __restrict__ lbias,
                                               float* __restrict__ out, int batch) {
    __shared__ float lds[8 * 256];
    const int wid  = threadIdx.x >> 5;
    const int lane = threadIdx.x & 31;
    const int b    = blockIdx.x * 8 + wid;
    if (b >= batch) return;               // wave-uniform guard
    // wave-uniform batch id in an SGPR so all PRNG math lowers to SALU
    const unsigned ub = (unsigned)__builtin_amdgcn_readfirstlane(b);
    float* S = lds + wid * 256;           // private 16x16 f32 state tile
    const int half = lane >> 4;
    const int n    = lane & 15;

    // amplitude-encode: S[row=hi nibble][col=lo nibble] = x[b, i]
    const float* xb = x + (size_t)b * 256;
#pragma unroll
    for (int k = 0; k < 8; k++) S[lane + 32 * k] = xb[lane + 32 * k];
    __builtin_amdgcn_wave_barrier();

    unsigned ev = 0x777u;
    for (int m = 0; m < NLAYERS; m++) {
        const float* Ahi = Rall + (m * 2 + 0) * 256;   // R_hi row-major (M,K)
        const float* Blo = Rall + (m * 2 + 1) * 256;   // R_lo row-major (N,K)

        // ---- GEMM1: D = R_hi * S  (4 chained WMMA f32 16x16x4) ----
        v8f d = {0, 0, 0, 0, 0, 0, 0, 0};
#pragma unroll
        for (int kk = 0; kk < 4; kk++) {
            int kb = 4 * kk + 2 * half;
            v2f a = *(const v2f*)(Ahi + n * 16 + kb);          // A(M=n, K=kb..kb+1)
            v2f bf; bf.x = S[kb * 16 + n]; bf.y = S[kb * 16 + 16 + n];  // B(K,N=n)
            d = wmma4(a, bf, d);
        }
#pragma unroll
        for (int v = 0; v < 8; v++) S[(v + 8 * half) * 16 + n] = d[v];
        __builtin_amdgcn_wave_barrier();

        // ---- GEMM2: S = D * R_lo^T ; B(k,n) = Blo[n*16+k] ----
        v8f d2 = {0, 0, 0, 0, 0, 0, 0, 0};
#pragma unroll
        for (int kk = 0; kk < 4; kk++) {
            int kb = 4 * kk + 2 * half;
            v2f a; a.x = S[n * 16 + kb]; a.y = S[n * 16 + kb + 1];
            v2f bf = *(const v2f*)(Blo + n * 16 + kb);
            d2 = wmma4(a, bf, d2);
        }
        __builtin_amdgcn_wave_barrier();
#pragma unroll
        for (int v = 0; v < 8; v++) S[(v + 8 * half) * 16 + n] = d2[v];
        __builtin_amdgcn_wave_barrier();

        // ---- per-qubit X/Z channel after each RY (commute across qubits) ----
        for (int q = 0; q < 8; q++)
            ev = xz_event(S, lane, 7 - q, T_P1Q, ev, ub);

        if (m < MLAY) {
            // idle Z dephasing
            for (int q = 0; q < 8; q++) {
                unsigned hz = hmix(ev++, ub);
                if (hz < T_PIDLE) gather_signed(S, lane, 0, 1 << (7 - q), 1);
            }
            // forward CNOT ring with 2q noise
            for (int q = 0; q < 8; q++) {
                int bc = 7 - q, bt = 7 - ((q + 1) & 7);
                apply_cnot(S, lane, bc, bt);
                ev = xz_event(S, lane, bc, T_P2Q1Q, ev, ub);
                ev = xz_event(S, lane, bt, T_P2Q1Q, ev, ub);
                ev = pair_event(S, lane, (1 << bc) | (1 << bt), T_P2QPR, ev, ub);
            }
        }
    }
    __builtin_amdgcn_wave_barrier();

    // ---- marginal over B qubits (low 3 bits of column), depol, readout, log ----
    if (lane < 8) {
        float p = 0.f;
#pragma unroll
        for (int r = 0; r < 16; r++) {
            float a0 = S[r * 16 + lane];
            float a1 = S[r * 16 + lane + 8];
            p += a0 * a0 + a1 * a1;
        }
        p = (1.0f - FDEPOL) * p + (FDEPOL / 8.0f);
#pragma unroll
        for (int bb = 0; bb < 3; bb++) {
            S[lane] = p;                                   // in-wave LDS exchange
            __builtin_amdgcn_wave_barrier();
            p = (1.0f - READOUT_C) * p + READOUT_C * S[lane ^ (1 << bb)];
            __builtin_amdgcn_wave_barrier();
        }
        out[(size_t)b * 8 + lane] = beta[0] * __logf(p + 1e-10f) + lbias[lane];
    }
}

// ---------------------------------------------------------------------------
extern "C" void kernel_launch(void* const* d_in, const int* in_sizes, int n_in,
                              void* d_out, int out_size, void* d_ws, size_t ws_size,
                              hipStream_t stream) {
    (void)n_in; (void)out_size; (void)ws_size;
    const float* x     = (const float*)d_in[0];   // (16384, 256)
    const float* theta = (const float*)d_in[1];   // (7, 8)
    const float* beta  = (const float*)d_in[2];   // scalar
    const float* lbias = (const float*)d_in[3];   // (8,)
    float* Rall = (float*)d_ws;                   // 14 x 16 x 16 f32 = 14 KB

    int batch = in_sizes[0] / 256;

    qae_prep<<<1, 256, 0, stream>>>(theta, Rall);
    qae_sim<<<(batch + 7) / 8, 256, 0, stream>>>(x, Rall, beta, lbias,
                                                 (float*)d_out, batch);
}